// MultiHeadAttentionWithKVCache_31980326486735
// MI455X (gfx1250) — compile-verified
//
#include <hip/hip_runtime.h>
#include <math.h>

typedef __attribute__((ext_vector_type(2))) float v2f;
typedef __attribute__((ext_vector_type(8))) float v8f;

#define D_MODEL   2048
#define NUM_HEADS 16
#define HEAD_DIM  128
#define BATCH     32
#define CACHE_LEN 4096
#define NEW_LEN   (CACHE_LEN + 1)          // 4097
#define CHUNK     512
#define NCHUNK    9                         // ceil(4097/512)

// ---------------------------------------------------------------------------
// GEMM: Y[32,2048] = X[32,2048] @ W[2048,2048] + b   via V_WMMA_F32_16X16X4_F32
// mode 0: Y stored row-major [32][2048]
// mode 1: Y scattered into cache row 4096: Y -> out[((m*16+h)*4097+4096)*128+d]
// Block = 256 threads (8 waves). Wave w: m-tile = w&1 (M=32 -> 2 tiles),
// n-subtile = w>>2? no: w>>1 (4 subtiles of 16 -> 64 cols per block).
// Grid.x = 2048/64 = 32 blocks. All tiles full => EXEC all ones at the WMMA.
// ---------------------------------------------------------------------------
__global__ __launch_bounds__(256)
void proj_wmma_f32(const float* __restrict__ X, const float* __restrict__ W,
                   const float* __restrict__ bias, float* __restrict__ Y, int mode)
{
    const int tid  = threadIdx.x;
    const int wave = tid >> 5;
    const int lane = tid & 31;
    const int hi   = lane >> 4;        // 0: lanes 0-15, 1: lanes 16-31
    const int ln   = lane & 15;

    const int m_base = (wave & 1) * 16;
    const int n0     = blockIdx.x * 64 + (wave >> 1) * 16;
    const int m      = m_base + ln;    // A-matrix row owned by this lane

    const float* xrow = X + (size_t)m * D_MODEL;
    const float* wcol = W + n0 + ln;

    v8f c = {};
    for (int k = 0; k < D_MODEL; k += 4) {
        // A 16x4: VGPR0 = K(0|2), VGPR1 = K(1|3) depending on lane half
        v2f a;
        a.x = xrow[k + 2 * hi];
        a.y = xrow[k + 2 * hi + 1];
        // B 4x16: VGPR0 = row K(0|2), VGPR1 = row K(1|3); N striped over 16 lanes
        v2f b;
        b.x = wcol[(size_t)(k + 2 * hi) * D_MODEL];
        b.y = wcol[(size_t)(k + 2 * hi + 1) * D_MODEL];
        c = __builtin_amdgcn_wmma_f32_16x16x4_f32(
                /*neg_a=*/false, a, /*neg_b=*/false, b,
                /*c_mod=*/(short)0, c, /*reuse_a=*/false, /*reuse_b=*/false);
    }

    // D 16x16: lane ln -> column n0+ln ; VGPR r -> row m_base + r + 8*hi
    const int  n  = n0 + ln;
    const float bv = bias[n];
    #pragma unroll
    for (int r = 0; r < 8; ++r) {
        const int   mm  = m_base + r + 8 * hi;
        const float val = c[r] + bv;
        if (mode == 0) {
            Y[(size_t)mm * D_MODEL + n] = val;
        } else {
            const int h = n >> 7, d = n & 127;
            Y[(((size_t)(mm * NUM_HEADS + h)) * NEW_LEN + CACHE_LEN) * HEAD_DIM + d] = val;
        }
    }
}

// ---------------------------------------------------------------------------
// Fused cache-concat + flash-decode chunk pass.
// Block = 256 threads (8 waves); grid = (NCHUNK, B*H).
// Each wave owns one key row per iteration: lane l holds dims [4l,4l+4).
// K/V elements are read from prev cache, written to the output cache (the
// concat), and consumed for scores / weighted-V in the same pass.
// Emits per-chunk partials (m, l, acc[128]) for a LSE reduce.
// ---------------------------------------------------------------------------
__global__ __launch_bounds__(256)
void attn_chunk(const float* __restrict__ Q,
                const float* __restrict__ prevK, const float* __restrict__ prevV,
                float* __restrict__ outK, float* __restrict__ outV,
                float* __restrict__ Pacc, float* __restrict__ Pml)
{
    const int c    = blockIdx.x;
    const int bh   = blockIdx.y;            // b*16 + h
    const int b    = bh >> 4, h = bh & 15;
    const int tid  = threadIdx.x;
    const int wave = tid >> 5;
    const int lane = tid & 31;

    __shared__ float qs[HEAD_DIM];
    __shared__ float sc[CHUNK];
    __shared__ float red[8];
    __shared__ __align__(16) float part[8][HEAD_DIM];

    const int k0 = c * CHUNK;
    const int nk = min(CHUNK, NEW_LEN - k0);

    if (tid < HEAD_DIM)
        qs[tid] = Q[(size_t)b * D_MODEL + h * HEAD_DIM + tid] * 0.08838834764831845f; // 1/sqrt(128)
    __syncthreads();

    const size_t oldBase = (size_t)bh * CACHE_LEN * HEAD_DIM;
    const size_t newBase = (size_t)bh * NEW_LEN   * HEAD_DIM;

    // ---- score pass (fused with K concat) ----
    for (int kk = wave; kk < nk; kk += 8) {
        const int key = k0 + kk;
        float4 kv;
        if (key < CACHE_LEN) {
            kv = *(const float4*)(prevK + oldBase + (size_t)key * HEAD_DIM + lane * 4);
            *(float4*)(outK + newBase + (size_t)key * HEAD_DIM + lane * 4) = kv;
        } else {
            kv = *(const float4*)(outK + newBase + (size_t)key * HEAD_DIM + lane * 4);
        }
        float p = kv.x * qs[4 * lane] + kv.y * qs[4 * lane + 1]
                + kv.z * qs[4 * lane + 2] + kv.w * qs[4 * lane + 3];
        #pragma unroll
        for (int off = 16; off; off >>= 1) p += __shfl_xor(p, off, 32);
        if (lane == 0) sc[kk] = p;
    }
    __syncthreads();

    // ---- chunk max ----
    float m = -INFINITY;
    for (int i = tid; i < nk; i += 256) m = fmaxf(m, sc[i]);
    #pragma unroll
    for (int off = 16; off; off >>= 1) m = fmaxf(m, __shfl_xor(m, off, 32));
    if (lane == 0) red[wave] = m;
    __syncthreads();
    float mb = red[0];
    #pragma unroll
    for (int w = 1; w < 8; ++w) mb = fmaxf(mb, red[w]);

    // ---- exp + chunk sum (sc becomes weights) ----
    float lsum = 0.f;
    for (int i = tid; i < nk; i += 256) {
        float e = __expf(sc[i] - mb);
        sc[i] = e;
        lsum += e;
    }
    #pragma unroll
    for (int off = 16; off; off >>= 1) lsum += __shfl_xor(lsum, off, 32);
    __syncthreads();                       // protect red reuse + finish sc writes
    if (lane == 0) red[wave] = lsum;
    __syncthreads();
    float lb = 0.f;
    #pragma unroll
    for (int w = 0; w < 8; ++w) lb += red[w];

    // ---- weighted-V pass (fused with V concat) ----
    float4 acc = make_float4(0.f, 0.f, 0.f, 0.f);
    for (int kk = wave; kk < nk; kk += 8) {
        const int key = k0 + kk;
        float4 vv;
        if (key < CACHE_LEN) {
            vv = *(const float4*)(prevV + oldBase + (size_t)key * HEAD_DIM + lane * 4);
            *(float4*)(outV + newBase + (size_t)key * HEAD_DIM + lane * 4) = vv;
        } else {
            vv = *(const float4*)(outV + newBase + (size_t)key * HEAD_DIM + lane * 4);
        }
        const float w = sc[kk];
        acc.x += w * vv.x; acc.y += w * vv.y; acc.z += w * vv.z; acc.w += w * vv.w;
    }
    *(float4*)&part[wave][lane * 4] = acc;
    __syncthreads();

    if (tid < HEAD_DIM) {
        float s = 0.f;
        #pragma unroll
        for (int w = 0; w < 8; ++w) s += part[w][tid];
        Pacc[((size_t)bh * NCHUNK + c) * HEAD_DIM + tid] = s;
    }
    if (tid == 0) {
        Pml[((size_t)bh * NCHUNK + c) * 2 + 0] = mb;
        Pml[((size_t)bh * NCHUNK + c) * 2 + 1] = lb;
    }
}

// ---------------------------------------------------------------------------
// Log-sum-exp reduce over chunks -> attn activations [32][2048]
// ---------------------------------------------------------------------------
__global__ __launch_bounds__(128)
void attn_reduce(const float* __restrict__ Pacc, const float* __restrict__ Pml,
                 float* __restrict__ attn)
{
    const int bh = blockIdx.x;
    const int d  = threadIdx.x;

    float M = -INFINITY;
    #pragma unroll
    for (int c = 0; c < NCHUNK; ++c)
        M = fmaxf(M, Pml[((size_t)bh * NCHUNK + c) * 2]);

    float L = 0.f, a = 0.f;
    #pragma unroll
    for (int c = 0; c < NCHUNK; ++c) {
        const float mm = Pml[((size_t)bh * NCHUNK + c) * 2 + 0];
        const float ll = Pml[((size_t)bh * NCHUNK + c) * 2 + 1];
        const float f  = __expf(mm - M);
        L += ll * f;
        a += Pacc[((size_t)bh * NCHUNK + c) * HEAD_DIM + d] * f;
    }
    const int b = bh >> 4, h = bh & 15;
    attn[(size_t)b * D_MODEL + h * HEAD_DIM + d] = a / L;
}

// ---------------------------------------------------------------------------
extern "C" void kernel_launch(void* const* d_in, const int* in_sizes, int n_in,
                              void* d_out, int out_size, void* d_ws, size_t ws_size,
                              hipStream_t stream)
{
    const float* x     = (const float*)d_in[0];
    const float* prevK = (const float*)d_in[1];
    const float* prevV = (const float*)d_in[2];
    const float* Wq    = (const float*)d_in[3];
    const float* bq    = (const float*)d_in[4];
    const float* Wk    = (const float*)d_in[5];
    const float* bk    = (const float*)d_in[6];
    const float* Wv    = (const float*)d_in[7];
    const float* bv    = (const float*)d_in[8];
    const float* Wo    = (const float*)d_in[9];
    const float* bo    = (const float*)d_in[10];

    float* out  = (float*)d_out;                               // [32,1,2048]
    float* outK = out  + (size_t)BATCH * D_MODEL;              // [32,16,4097,128]
    float* outV = outK + (size_t)BATCH * NUM_HEADS * NEW_LEN * HEAD_DIM;

    float* ws   = (float*)d_ws;
    float* Qws  = ws;                                          // 65536 floats
    float* attn = Qws  + (size_t)BATCH * D_MODEL;              // 65536 floats
    float* Pml  = attn + (size_t)BATCH * D_MODEL;              // 512*9*2
    float* Pacc = Pml  + (size_t)BATCH * NUM_HEADS * NCHUNK * 2; // 512*9*128

    // QKV projections (WMMA f32). K/V new rows land directly in the output caches.
    proj_wmma_f32<<<dim3(D_MODEL / 64), dim3(256), 0, stream>>>(x, Wq, bq, Qws,  0);
    proj_wmma_f32<<<dim3(D_MODEL / 64), dim3(256), 0, stream>>>(x, Wk, bk, outK, 1);
    proj_wmma_f32<<<dim3(D_MODEL / 64), dim3(256), 0, stream>>>(x, Wv, bv, outV, 1);

    // Fused cache-concat + flash-decode partials, then LSE reduce.
    attn_chunk<<<dim3(NCHUNK, BATCH * NUM_HEADS), dim3(256), 0, stream>>>(
        Qws, prevK, prevV, outK, outV, Pacc, Pml);
    attn_reduce<<<dim3(BATCH * NUM_HEADS), dim3(128), 0, stream>>>(Pacc, Pml, attn);

    // Output projection (WMMA f32) -> final out
    proj_wmma_f32<<<dim3(D_MODEL / 64), dim3(256), 0, stream>>>(attn, Wo, bo, out, 0);
}